// TopKTopPSampler_41085657153656
// MI455X (gfx1250) — compile-verified
//
#include <hip/hip_runtime.h>
#include <stdint.h>

// Sort-free top-k/top-p logit masking for [B,V]=[256,128000] f32.
// One 256-thread block per row.
//   Pass 1 (HBM): 12-bit key histogram + row max, streamed through LDS with
//                 gfx1250 async-to-LDS loads (ASYNCcnt), depth-4 pipeline.
//   Pass 2 (L2):  refine next 12 bits AND compact all possible candidates
//                 (key >= coarse-bin start, <= ~2K elements) into LDS.
//   LDS-only:     exact k-th key, exp-mass Z, weighted 12/12/8 radix descent
//                 for the exact top-p threshold key.
//   Pass 3: non-temporal L2-bypass read + HBM write of the masked row.
// HBM-critical traffic ~= 131MB read + 131MB write => ~11us @ 23.3 TB/s.

#define ROW_V     128000
#define ROW_V4    (ROW_V / 4)        // 32000 float4 per row
#define NTHREADS  256
#define NTILES    (ROW_V4 / NTHREADS) // 125 tiles of 256 float4
#define NBINS     4096
#define NCOPIES   2                  // privatized LDS histograms (32KB)
#define CAND_MAX  4096               // candidate value buffer (16KB)
#define PIPE      4                  // async staging depth (16KB, aliases cand)

typedef float f4v __attribute__((ext_vector_type(4)));

#if defined(__gfx1250__) && __has_builtin(__builtin_amdgcn_global_load_async_to_lds_b128)
#define HAVE_ASYNC_LDS 1
// Probe-confirmed: param 0 is an AS1 pointer to a 16B int vector.
typedef int v4i_b128 __attribute__((vector_size(16)));
typedef __attribute__((address_space(1))) v4i_b128 GV4;
typedef __attribute__((address_space(3))) v4i_b128 LV4;
#if __has_builtin(__builtin_amdgcn_s_wait_asynccnt)
#define WAIT_ASYNC(n) __builtin_amdgcn_s_wait_asynccnt(n)
#else
#define WAIT_ASYNC(n) asm volatile("s_wait_asynccnt %0" ::"n"(n) : "memory")
#endif
#endif

// Monotonic float -> uint32 key (ascending order preserved).
__device__ __forceinline__ uint32_t f2key(float f) {
    uint32_t u = __float_as_uint(f);
    return u ^ ((uint32_t)((int32_t)u >> 31) | 0x80000000u);
}

__device__ __forceinline__ float key2f(uint32_t k) {
    uint32_t u = (k & 0x80000000u) ? (k ^ 0x80000000u) : ~k;
    return __uint_as_float(u);
}

__global__ __launch_bounds__(NTHREADS)
void topk_topp_kernel(const float* __restrict__ logits,
                      const int*   __restrict__ kArr,
                      const float* __restrict__ pArr,
                      float*       __restrict__ out)
{
    __shared__ uint32_t hist[NCOPIES][NBINS];
    __shared__ __align__(16) float cand[CAND_MAX];  // also async staging in P1
    __shared__ uint32_t s_candCnt;
    __shared__ uint32_t s_maxKey;
    __shared__ uint32_t s_sel;     // evolving top-k key prefix
    __shared__ uint32_t s_rem;     // remaining rank inside selected bin
    __shared__ uint32_t s_kk;      // exact key of the k-th largest element
    __shared__ uint32_t s_pfx;     // evolving top-p key prefix
    __shared__ float    s_target;  // remaining target mass inside bin
    __shared__ uint32_t s_done;    // top-p resolved early (fallback)
    __shared__ uint32_t s_kp;      // final keep threshold key (>= s_kk)

    const int row = blockIdx.x;
    const int tid = threadIdx.x;
    const f4v* __restrict__ row4 =
        reinterpret_cast<const f4v*>(logits + (size_t)row * ROW_V);

    uint32_t* myH  = hist[(tid >= NTHREADS / 2) ? 1 : 0];
    float*    myHF = reinterpret_cast<float*>(myH);
    uint32_t* h0   = hist[0];
    uint32_t* h1   = hist[1];
    float*    h0F  = reinterpret_cast<float*>(h0);
    float*    h1F  = reinterpret_cast<float*>(h1);

    // ===== Phase 1: row max + count histogram of key[31:20] (HBM stream) ====
    for (int i = tid; i < NCOPIES * NBINS; i += NTHREADS) (&hist[0][0])[i] = 0u;
    if (tid == 0) s_maxKey = 0u;
    __syncthreads();

    uint32_t lmax = 0u;
#if HAVE_ASYNC_LDS
    {
        // Depth-4 async staging: each lane DMAs its own 16B/tile into LDS and
        // reads back only its own slot, so a per-wave s_wait_asynccnt is the
        // only ordering required (no cross-thread barrier). 3 tiles stay in
        // flight behind the one being consumed to cover HBM latency.
        f4v* stage = reinterpret_cast<f4v*>(cand);   // PIPE*256*16B = 16KB
        #pragma unroll
        for (int t0 = 0; t0 < PIPE - 1; ++t0)
            __builtin_amdgcn_global_load_async_to_lds_b128(
                (GV4*)(row4 + (size_t)t0 * NTHREADS + tid),
                (LV4*)&stage[t0 * NTHREADS + tid], 0, 0);
        for (int t = 0; t < NTILES; ++t) {
            if (t + PIPE - 1 < NTILES) {
                __builtin_amdgcn_global_load_async_to_lds_b128(
                    (GV4*)(row4 + (size_t)(t + PIPE - 1) * NTHREADS + tid),
                    (LV4*)&stage[((t + PIPE - 1) & (PIPE - 1)) * NTHREADS + tid],
                    0, 0);
                WAIT_ASYNC(PIPE - 1);
            } else {
                WAIT_ASYNC(0);     // drain tail (constant-immediate wait)
            }
            asm volatile("" ::: "memory");  // keep the LDS read below the wait
            f4v v = stage[(t & (PIPE - 1)) * NTHREADS + tid];
            uint32_t a = f2key(v.x), b = f2key(v.y), c = f2key(v.z), d = f2key(v.w);
            uint32_t m01 = a > b ? a : b, m23 = c > d ? c : d;
            uint32_t m = m01 > m23 ? m01 : m23;
            lmax = m > lmax ? m : lmax;
            atomicAdd(&myH[a >> 20], 1u);
            atomicAdd(&myH[b >> 20], 1u);
            atomicAdd(&myH[c >> 20], 1u);
            atomicAdd(&myH[d >> 20], 1u);
        }
    }
#else
    for (int i = tid; i < ROW_V4; i += NTHREADS) {
        if (i + NTHREADS < ROW_V4)
            __builtin_prefetch(row4 + i + NTHREADS, 0, 1);
        f4v v = row4[i];
        uint32_t a = f2key(v.x), b = f2key(v.y), c = f2key(v.z), d = f2key(v.w);
        uint32_t m01 = a > b ? a : b, m23 = c > d ? c : d;
        uint32_t m = m01 > m23 ? m01 : m23;
        lmax = m > lmax ? m : lmax;
        atomicAdd(&myH[a >> 20], 1u);
        atomicAdd(&myH[b >> 20], 1u);
        atomicAdd(&myH[c >> 20], 1u);
        atomicAdd(&myH[d >> 20], 1u);
    }
#endif
    atomicMax(&s_maxKey, lmax);
    __syncthreads();
    for (int b = tid; b < NBINS; b += NTHREADS) h0[b] += h1[b];
    __syncthreads();
    if (tid == 0) {
        uint32_t kk = (uint32_t)kArr[row];   // 1..1024
        uint32_t cum = 0; int sel = 0;
        for (int d = NBINS - 1; d >= 0; --d) {
            uint32_t c = h0[d];
            if (cum + c >= kk) { sel = d; break; }
            cum += c;
        }
        s_sel = (uint32_t)sel << 20;
        s_rem = kk - cum;
        s_candCnt = 0u;
    }
    __syncthreads();

    // ===== Phase 2 (L2): refine key[19:8] AND compact candidates ============
    const uint32_t pfx1 = s_sel;
    for (int i = tid; i < NCOPIES * NBINS; i += NTHREADS) (&hist[0][0])[i] = 0u;
    __syncthreads();
    for (int i = tid; i < ROW_V4; i += NTHREADS) {
        if (i + NTHREADS < ROW_V4)
            __builtin_prefetch(row4 + i + NTHREADS, 0, 1);  // global_prefetch_b8
        f4v v = row4[i];
        float x[4] = {v.x, v.y, v.z, v.w};
        #pragma unroll
        for (int j = 0; j < 4; ++j) {
            uint32_t kj = f2key(x[j]);
            if (kj >= pfx1) {
                uint32_t slot = atomicAdd(&s_candCnt, 1u);
                if (slot < CAND_MAX) cand[slot] = x[j];   // overflow: degrade
                if ((kj & 0xFFF00000u) == pfx1)
                    atomicAdd(&myH[(kj >> 8) & 0xFFFu], 1u);
            }
        }
    }
    __syncthreads();
    for (int b = tid; b < NBINS; b += NTHREADS) h0[b] += h1[b];
    __syncthreads();
    if (tid == 0) {
        uint32_t rem = s_rem, cum = 0; int sel = 0;
        for (int d = NBINS - 1; d >= 0; --d) {
            uint32_t c = h0[d];
            if (cum + c >= rem) { sel = d; break; }
            cum += c;
        }
        s_sel = pfx1 | ((uint32_t)sel << 8);
        s_rem = rem - cum;
    }
    __syncthreads();

    const int cn = (int)(s_candCnt < CAND_MAX ? s_candCnt : CAND_MAX);

    // ===== Phase 3 (LDS): refine key[7:0] -> exact k-th largest key =========
    const uint32_t pfx2 = s_sel;
    for (int i = tid; i < NCOPIES * NBINS; i += NTHREADS) (&hist[0][0])[i] = 0u;
    __syncthreads();
    for (int i = tid; i < cn; i += NTHREADS) {
        uint32_t kj = f2key(cand[i]);
        if ((kj & 0xFFFFFF00u) == pfx2) atomicAdd(&myH[kj & 0xFFu], 1u);
    }
    __syncthreads();
    for (int b = tid; b < 256; b += NTHREADS) h0[b] += h1[b];
    __syncthreads();
    if (tid == 0) {
        uint32_t rem = s_rem, cum = 0; int sel = 0;
        for (int d = 255; d >= 0; --d) {
            uint32_t c = h0[d];
            if (cum + c >= rem) { sel = d; break; }
            cum += c;
        }
        s_kk = pfx2 | (uint32_t)sel;
    }
    __syncthreads();

    // ===== Phase 4 (LDS): exp-mass histogram of survivors, key[31:20] =======
    const uint32_t Kk     = s_kk;
    const float    maxVal = key2f(s_maxKey);
    const float    pRow   = pArr[row];
    for (int i = tid; i < NCOPIES * NBINS; i += NTHREADS) (&hist[0][0])[i] = 0u;
    __syncthreads();
    for (int i = tid; i < cn; i += NTHREADS) {
        float x = cand[i];
        uint32_t kj = f2key(x);
        if (kj >= Kk) atomicAdd(&myHF[kj >> 20], expf(x - maxVal));
    }
    __syncthreads();
    for (int b = tid; b < NBINS; b += NTHREADS) h0F[b] += h1F[b];
    __syncthreads();
    if (tid == 0) {
        float Z = 0.f;                       // total kept mass
        for (int d = 0; d < NBINS; ++d) Z += h0F[d];
        float t = pRow * Z;                  // keep x iff mass-above(x) < t
        float cum = 0.f; int sel = -1;
        for (int d = NBINS - 1; d >= 0; --d) {
            float m = h0F[d];
            if (m > 0.f && cum + m >= t) { sel = d; break; }
            cum += m;
        }
        if (sel < 0) { s_done = 1u; s_kp = Kk; }    // rounding fallback
        else { s_done = 0u; s_pfx = (uint32_t)sel << 20; s_target = t - cum; }
    }
    __syncthreads();

    // ===== Phase 5 (LDS): weighted refine key[19:8] =========================
    uint32_t done  = s_done;
    uint32_t ppfx1 = s_pfx;
    for (int i = tid; i < NCOPIES * NBINS; i += NTHREADS) (&hist[0][0])[i] = 0u;
    __syncthreads();
    if (!done) {
        for (int i = tid; i < cn; i += NTHREADS) {
            float x = cand[i];
            uint32_t kj = f2key(x);
            if (kj >= Kk && (kj & 0xFFF00000u) == ppfx1)
                atomicAdd(&myHF[(kj >> 8) & 0xFFFu], expf(x - maxVal));
        }
    }
    __syncthreads();
    for (int b = tid; b < NBINS; b += NTHREADS) h0F[b] += h1F[b];
    __syncthreads();
    if (tid == 0 && !done) {
        float t = s_target, cum = 0.f; int sel = -1;
        for (int d = NBINS - 1; d >= 0; --d) {
            float m = h0F[d];
            if (m > 0.f && cum + m >= t) { sel = d; break; }
            cum += m;
        }
        if (sel < 0) { s_done = 1u; s_kp = Kk; }
        else { s_pfx = ppfx1 | ((uint32_t)sel << 8); s_target = t - cum; }
    }
    __syncthreads();

    // ===== Phase 6 (LDS): weighted refine key[7:0] -> exact keep key ========
    done = s_done;
    uint32_t ppfx2 = s_pfx;
    for (int i = tid; i < NCOPIES * NBINS; i += NTHREADS) (&hist[0][0])[i] = 0u;
    __syncthreads();
    if (!done) {
        for (int i = tid; i < cn; i += NTHREADS) {
            float x = cand[i];
            uint32_t kj = f2key(x);
            if (kj >= Kk && (kj & 0xFFFFFF00u) == ppfx2)
                atomicAdd(&myHF[kj & 0xFFu], expf(x - maxVal));
        }
    }
    __syncthreads();
    for (int b = tid; b < 256; b += NTHREADS) h0F[b] += h1F[b];
    __syncthreads();
    if (tid == 0 && !done) {
        float t = s_target, cum = 0.f; int sel = -1;
        for (int d = 255; d >= 0; --d) {
            float m = h0F[d];
            if (m > 0.f && cum + m >= t) { sel = d; break; }
            cum += m;
        }
        s_kp = (sel < 0) ? Kk : (ppfx2 | (uint32_t)sel);
    }
    __syncthreads();

    // ===== Phase 7: masked write-back, non-temporal (last use of logits) ====
    const uint32_t Kp = s_kp;                 // Kp >= Kk by construction
    const float NEGINF = __int_as_float(0xFF800000);
    f4v* __restrict__ out4 = reinterpret_cast<f4v*>(out + (size_t)row * ROW_V);
    for (int i = tid; i < ROW_V4; i += NTHREADS) {
        f4v v = __builtin_nontemporal_load(&row4[i]);
        f4v o;
        o.x = (f2key(v.x) >= Kp) ? v.x : NEGINF;
        o.y = (f2key(v.y) >= Kp) ? v.y : NEGINF;
        o.z = (f2key(v.z) >= Kp) ? v.z : NEGINF;
        o.w = (f2key(v.w) >= Kp) ? v.w : NEGINF;
        __builtin_nontemporal_store(o, &out4[i]);
    }
}

extern "C" void kernel_launch(void* const* d_in, const int* in_sizes, int n_in,
                              void* d_out, int out_size, void* d_ws, size_t ws_size,
                              hipStream_t stream) {
    (void)n_in; (void)d_ws; (void)ws_size; (void)out_size;
    const float* logits = (const float*)d_in[0];
    const int*   k      = (const int*)d_in[1];
    const float* p      = (const float*)d_in[2];
    float*       out    = (float*)d_out;
    const int B = in_sizes[1];   // one k entry per row
    topk_topp_kernel<<<dim3(B), dim3(NTHREADS), 0, stream>>>(logits, k, p, out);
}